// FWHT_26843545600267
// MI455X (gfx1250) — compile-verified
//
#include <hip/hip_runtime.h>
#include <hip/hip_bf16.h>

// ---------------------------------------------------------------------------
// FWHT (64, 1024, 512) f32, transform along axis 1 (n = 1024).
// Memory-bound (1.3 FLOP/byte -> ~11 us floor at 23.3 TB/s): single pass,
// full 1024-pt transform staged in LDS, read once / write once.
// H_1024 = (H_4 (x) I_256) (I_4 (x) H_16 (x) I_16) (I_64 (x) H_16),
//   i = 256*q + 16*p + l.
// H_16 stages -> v_wmma_f32_16x16x4_f32 (x4 chained over K, full f32 precision).
// Global <-> LDS via CDNA5 async-to-LDS ops (ASYNCcnt), no VGPR staging.
// ---------------------------------------------------------------------------

#define NB 64
#define NI 1024
#define ND 512
#define DT 16            // d-columns per block
#define RS 16            // LDS row stride in floats -> 1024*16*4 = 64 KB

#ifndef __has_builtin
#define __has_builtin(x) 0
#endif
#if __has_builtin(__builtin_amdgcn_wmma_f32_16x16x4_f32)
#define HAVE_WMMA_F32 1
#else
#define HAVE_WMMA_F32 0
#endif
#define USE_ASYNC_LDS 1

typedef float v2f __attribute__((ext_vector_type(2)));
typedef float v8f __attribute__((ext_vector_type(8)));

__device__ __forceinline__ float hsign(int m, int k) {
    return (__builtin_popcount(m & k) & 1) ? -1.0f : 1.0f;
}

// One wave applies D = H16 * T to a 16x16 tile living in LDS.
// Row k of the tile is at float index rowBase + k*rowStride + col (col 0..15).
// Fragment layouts per CDNA5 ISA 7.12.2 (32-bit, wave32):
//   A 16x4 : lane m=lane&15; vgpr v holds K = v + 2*(lane>=16)
//   B 4x16 : lane n=lane&15; vgpr v holds K = v + 2*(lane>=16)  (C/D-style striping)
//   D 16x16: lane n=lane&15; vgpr r holds M = r + 8*(lane>=16)
// All tile reads precede all tile writes in wave program order, and each tile
// is owned by exactly one wave, so no intra-phase synchronization is needed.
__device__ __forceinline__ void h16_tile(float* ls, int rowBase, int rowStride) {
    const int lane = threadIdx.x & 31;
    const int n    = lane & 15;             // col for B/D, row m for A
    const int kb   = (lane >> 4) << 1;      // 0 or 2
    const int rb   = (lane >> 4) << 3;      // 0 or 8

#if HAVE_WMMA_F32
    v8f acc = {};
#pragma unroll
    for (int c = 0; c < 4; ++c) {
        const int k0 = (c << 2) + kb;
        v2f a, bb;
        a.x  = hsign(n, k0);
        a.y  = hsign(n, k0 + 1);
        bb.x = ls[rowBase + (k0    ) * rowStride + n];
        bb.y = ls[rowBase + (k0 + 1) * rowStride + n];
        // 8 args: (neg_a, A, neg_b, B, c_mod, C, reuse_a, reuse_b)
        acc = __builtin_amdgcn_wmma_f32_16x16x4_f32(
            false, a, false, bb, (short)0, acc, false, false);
    }
#pragma unroll
    for (int r = 0; r < 8; ++r)
        ls[rowBase + (rb + r) * rowStride + n] = acc[r];
#else
    // Scalar fallback with the identical read/write footprint.
    float col[16];
#pragma unroll
    for (int k = 0; k < 16; ++k)
        col[k] = ls[rowBase + k * rowStride + n];
    float out[8];
#pragma unroll
    for (int r = 0; r < 8; ++r) {
        float s = 0.0f;
#pragma unroll
        for (int k = 0; k < 16; ++k) s += hsign(rb + r, k) * col[k];
        out[r] = s;
    }
#pragma unroll
    for (int r = 0; r < 8; ++r)
        ls[rowBase + (rb + r) * rowStride + n] = out[r];
#endif
}

__global__ __launch_bounds__(256)
void fwht1024_kernel(const float* __restrict__ x, float* __restrict__ y) {
    __shared__ float ls[NI * RS];           // 64 KB: 1024 rows x 16 d-cols

    const int tid  = threadIdx.x;
    const int wave = tid >> 5;              // 8 waves per block
    const int blk  = blockIdx.x;            // 2048 blocks
    const int b    = blk >> 5;              // batch
    const int dt   = blk & 31;              // d-tile
    const size_t gbase = (size_t)b * (NI * ND) + (size_t)dt * DT;

    // ---- Load 1024x16 tile: 4096 x b128, per-lane global->LDS async DMA.
    // Fully unrolled: 16 back-to-back async ops per thread, one wait at end.
#pragma unroll
    for (int it = 0; it < 16; ++it) {
        const int e  = tid + (it << 8);     // float4 index
        const int i  = e >> 2;              // row (transform axis)
        const int dq = (e & 3) << 2;        // d offset (floats)
        const float* gp = x + gbase + (size_t)i * ND + dq;
        const int loff  = i * RS + dq;
#if USE_ASYNC_LDS
        unsigned lo = (unsigned)(uintptr_t)&ls[loff];  // LDS byte offset (low 32 bits)
        asm volatile("global_load_async_to_lds_b128 %0, %1, off"
                     :: "v"(lo), "v"(gp) : "memory");
#else
        *(float4*)&ls[loff] = *(const float4*)gp;
#endif
    }
#if USE_ASYNC_LDS
    asm volatile("s_wait_asynccnt 0" ::: "memory");
#endif
    __syncthreads();

    // ---- Phase 1: I_64 (x) H_16 (transform over l). 64 tiles, 8 per wave.
    // Compile-time trip count: no EXEC-mask loop, WMMA runs with EXEC all-1s.
#pragma unroll 2
    for (int j = 0; j < 8; ++j)
        h16_tile(ls, (wave + (j << 3)) * (16 * RS), RS);
    __syncthreads();

    // ---- Phase 2: I_4 (x) H_16 (x) I_16 (transform over p). Tile per (q,l).
#pragma unroll 2
    for (int j = 0; j < 8; ++j) {
        const int t = wave + (j << 3);
        const int q = t >> 4, l = t & 15;
        h16_tile(ls, (256 * q + l) * RS, 16 * RS);
    }
    __syncthreads();

    // ---- Phase 3: H_4 (x) I_256 (butterflies over q in VALU) ----
    {
        const int l = (tid >> 4) & 15;
        const int d = tid & 15;
#pragma unroll
        for (int p = 0; p < 16; ++p) {
            const int a0 = (16 * p + l) * RS + d;
            const float v0 = ls[a0];
            const float v1 = ls[a0 + 256 * RS];
            const float v2 = ls[a0 + 512 * RS];
            const float v3 = ls[a0 + 768 * RS];
            const float t0 = v0 + v1, t1 = v0 - v1;
            const float t2 = v2 + v3, t3 = v2 - v3;
            ls[a0]            = t0 + t2;
            ls[a0 + 256 * RS] = t1 + t3;
            ls[a0 + 512 * RS] = t0 - t2;
            ls[a0 + 768 * RS] = t1 - t3;
        }
    }
    __syncthreads();

    // ---- Store tile: per-lane LDS->global async DMA, fully unrolled ----
#pragma unroll
    for (int it = 0; it < 16; ++it) {
        const int e  = tid + (it << 8);
        const int i  = e >> 2;
        const int dq = (e & 3) << 2;
        float* gp = y + gbase + (size_t)i * ND + dq;
        const int loff = i * RS + dq;
#if USE_ASYNC_LDS
        unsigned lo = (unsigned)(uintptr_t)&ls[loff];
        asm volatile("global_store_async_from_lds_b128 %0, %1, off"
                     :: "v"(gp), "v"(lo) : "memory");
#else
        *(float4*)gp = *(const float4*)&ls[loff];
#endif
    }
#if USE_ASYNC_LDS
    asm volatile("s_wait_asynccnt 0" ::: "memory");
#endif
}

extern "C" void kernel_launch(void* const* d_in, const int* in_sizes, int n_in,
                              void* d_out, int out_size, void* d_ws, size_t ws_size,
                              hipStream_t stream) {
    (void)in_sizes; (void)n_in; (void)out_size; (void)d_ws; (void)ws_size;
    const float* x = (const float*)d_in[0];
    float* y = (float*)d_out;
    const int blocks = NB * (ND / DT);      // 64 * 32 = 2048
    fwht1024_kernel<<<blocks, 256, 0, stream>>>(x, y);
}